// LennardJones_7138235646413
// MI455X (gfx1250) — compile-verified
//
#include <hip/hip_runtime.h>
#include <stdint.h>

#define BLOCK 256
#define TAB_STRIDE 64   // floats between the 3 tables in LDS (supports NS<=8)
#define PPT 4           // pairs per thread

typedef int v4i __attribute__((ext_vector_type(4)));

// Guaranteed-native fp32 global atomic add, no return value (STOREcnt).
__device__ __forceinline__ void atomic_add_f32_nr(float* addr, float val) {
    asm volatile("global_atomic_add_f32 %0, %1, off"
                 :: "v"((uint64_t)(uintptr_t)addr), "v"(val)
                 : "memory");
}

__device__ __forceinline__ void lj_pair_one(
    int i, int j, float fx, float fy, float fz,
    const float* __restrict__ pos, const int* __restrict__ species,
    const float* s_tab, int ns,
    float c00, float c01, float c02,
    float c10, float c11, float c12,
    float c20, float c21, float c22,
    float* __restrict__ energy)
{
    // Gathers served out of the 192 MB L2 (positions 6 MB, species 2 MB).
    float pix = pos[3 * i + 0], piy = pos[3 * i + 1], piz = pos[3 * i + 2];
    float pjx = pos[3 * j + 0], pjy = pos[3 * j + 1], pjz = pos[3 * j + 2];
    int si = species[i];
    int sj = species[j];

    // d = r_j - r_i + S @ cell   (S is a row vector)
    float dx = pjx - pix + fx * c00 + fy * c10 + fz * c20;
    float dy = pjy - piy + fx * c01 + fy * c11 + fz * c21;
    float dz = pjz - piz + fx * c02 + fy * c12 + fz * c22;

    int idx = si * ns + sj;
    float sg = s_tab[0 * TAB_STRIDE + idx];
    float ep = s_tab[1 * TAB_STRIDE + idx];
    float sh = s_tab[2 * TAB_STRIDE + idx];

    float r2  = dx * dx + dy * dy + dz * dz;
    float r6  = r2 * r2 * r2;
    float s3  = sg * sg * sg;
    float s6  = s3 * s3;
    float sr6  = s6 / r6;
    float sr12 = sr6 * sr6;
    float eh = 0.5f * (4.0f * ep * (sr12 - sr6) - sh);

    // Native fp32 L2 atomics (no return, STOREcnt; s_endpgm's implicit
    // wait-idle covers completion).
    atomic_add_f32_nr(energy + i, eh);
    atomic_add_f32_nr(energy + j, eh);
}

__global__ __launch_bounds__(BLOCK) void lj_pair_kernel(
    const float* __restrict__ pos,        // [N,3]
    const float* __restrict__ cell,       // [3,3]
    const float* __restrict__ sigma_tab,  // [NS,NS]
    const float* __restrict__ eps_tab,    // [NS,NS]
    const float* __restrict__ shift_tab,  // [NS,NS]
    const int*   __restrict__ species,    // [N]
    const int*   __restrict__ pair_i,     // [P]
    const int*   __restrict__ pair_j,     // [P]
    const int*   __restrict__ shifts,     // [P,3]
    float*       __restrict__ energy,     // [N] (pre-zeroed)
    int P, int ns, int ns2)
{
    __shared__ float s_tab[3 * TAB_STRIDE];

    // ---- Stage the three NSxNS parameter tables into LDS via the async
    // ---- LDS-DMA engine (gfx1250 GLOBAL_LOAD_ASYNC_TO_LDS_B32, ASYNCcnt).
    {
        int t = threadIdx.x;
        const float* src = nullptr;
        float* dst = nullptr;
        if (t < ns2) {
            src = sigma_tab + t;            dst = &s_tab[0 * TAB_STRIDE + t];
        } else if (t < 2 * ns2) {
            src = eps_tab + (t - ns2);      dst = &s_tab[1 * TAB_STRIDE + (t - ns2)];
        } else if (t < 3 * ns2) {
            src = shift_tab + (t - 2*ns2);  dst = &s_tab[2 * TAB_STRIDE + (t - 2*ns2)];
        }
        if (src) {
            // Low 32 bits of a generic LDS pointer are the LDS byte offset.
            uint32_t lds_off = (uint32_t)(uintptr_t)dst;
            uint64_t gaddr   = (uint64_t)(uintptr_t)src;
            asm volatile("global_load_async_to_lds_b32 %0, %1, off"
                         :: "v"(lds_off), "v"(gaddr) : "memory");
        }
        asm volatile("s_wait_asynccnt 0" ::: "memory");
    }
    __syncthreads();

    // Uniform 3x3 cell -> scalar loads.
    float c00 = cell[0], c01 = cell[1], c02 = cell[2];
    float c10 = cell[3], c11 = cell[4], c12 = cell[5];
    float c20 = cell[6], c21 = cell[7], c22 = cell[8];

    int pbase = (blockIdx.x * BLOCK + threadIdx.x) * PPT;
    if (pbase >= P) return;

    if (pbase + PPT <= P) {
        // ---- Fast path: the whole 20 B/pair stream arrives as five aligned
        // ---- global_load_b128 NT loads per 4 pairs (no cacheline straddle,
        // ---- no L2 pollution on a 320 MB use-once stream).
        v4i vi = __builtin_nontemporal_load((const v4i*)(pair_i + pbase));
        v4i vj = __builtin_nontemporal_load((const v4i*)(pair_j + pbase));
        const v4i* sp = (const v4i*)(shifts + 3 * pbase);   // 48t bytes: 16B aligned
        v4i s0 = __builtin_nontemporal_load(sp + 0);
        v4i s1 = __builtin_nontemporal_load(sp + 1);
        v4i s2 = __builtin_nontemporal_load(sp + 2);

        int ii[PPT] = { vi.x, vi.y, vi.z, vi.w };
        int jj[PPT] = { vj.x, vj.y, vj.z, vj.w };
        float fx[PPT] = { (float)s0.x, (float)s0.w, (float)s1.z, (float)s2.y };
        float fy[PPT] = { (float)s0.y, (float)s1.x, (float)s1.w, (float)s2.z };
        float fz[PPT] = { (float)s0.z, (float)s1.y, (float)s2.x, (float)s2.w };

#pragma unroll
        for (int k = 0; k < PPT; ++k) {
            lj_pair_one(ii[k], jj[k], fx[k], fy[k], fz[k],
                        pos, species, s_tab, ns,
                        c00, c01, c02, c10, c11, c12, c20, c21, c22, energy);
        }
    } else {
        // ---- Tail: scalar pairs.
        for (int p = pbase; p < P; ++p) {
            int i = pair_i[p];
            int j = pair_j[p];
            float fx = (float)shifts[3 * p + 0];
            float fy = (float)shifts[3 * p + 1];
            float fz = (float)shifts[3 * p + 2];
            lj_pair_one(i, j, fx, fy, fz,
                        pos, species, s_tab, ns,
                        c00, c01, c02, c10, c11, c12, c20, c21, c22, energy);
        }
    }
}

__global__ __launch_bounds__(BLOCK) void lj_zero_kernel(float* __restrict__ out, int n) {
    int t = blockIdx.x * BLOCK + threadIdx.x;
    if (t < n) out[t] = 0.0f;
}

extern "C" void kernel_launch(void* const* d_in, const int* in_sizes, int n_in,
                              void* d_out, int out_size, void* d_ws, size_t ws_size,
                              hipStream_t stream) {
    const float* pos       = (const float*)d_in[0];
    const float* cell      = (const float*)d_in[1];
    const float* sigma_tab = (const float*)d_in[2];
    const float* eps_tab   = (const float*)d_in[3];
    const float* shift_tab = (const float*)d_in[4];
    const int*   species   = (const int*)d_in[5];
    const int*   pair_i    = (const int*)d_in[6];
    const int*   pair_j    = (const int*)d_in[7];
    const int*   shifts    = (const int*)d_in[8];

    int N   = in_sizes[5];
    int P   = in_sizes[6];
    int ns2 = in_sizes[2];
    int ns  = 1;
    while (ns * ns < ns2) ++ns;

    float* energy = (float*)d_out;

    lj_zero_kernel<<<(N + BLOCK - 1) / BLOCK, BLOCK, 0, stream>>>(energy, N);

    int pairs_per_block = BLOCK * PPT;
    lj_pair_kernel<<<(P + pairs_per_block - 1) / pairs_per_block, BLOCK, 0, stream>>>(
        pos, cell, sigma_tab, eps_tab, shift_tab, species,
        pair_i, pair_j, shifts, energy, P, ns, ns2);
}